// BarycentricCoordinates_70540542869532
// MI455X (gfx1250) — compile-verified
//
#include <hip/hip_runtime.h>
#include <math.h>

#define NVERT   6144
#define KNN     32
#define NTPL    40      // 5 * 8
#define EPSF    1e-12f
#define BIGF    1e30f

typedef float v2f __attribute__((ext_vector_type(2)));
typedef float v8f __attribute__((ext_vector_type(8)));

// ---------------- wave32 helpers ----------------
__device__ __forceinline__ float wave_sum32(float v) {
#pragma unroll
    for (int m = 16; m >= 1; m >>= 1) v += __shfl_xor(v, m, 32);
    return v;
}

// ---------------- Kernel 1: fused WMMA distance tiles + top-33 selection ---
// grid: 2 * (NVERT/16) blocks, 256 threads (8 waves). Each block owns 16 rows.
// d2 tile (16x16) computed by V_WMMA_F32_16X16X4_F32:
//   A row m  = ( x_m,  y_m,  z_m, |v_m|^2 )          (K = 4)
//   B col n  = (-2x_n,-2y_n,-2z_n, 1 )
//   C[m][n]  = |v_n|^2  (broadcast over rows per lane-column)
//   D[m][n]  = |v_m|^2 - 2 v_m.v_n + |v_n|^2 = d2(m,n)
__global__ __launch_bounds__(256)
void knn_kernel(const float* __restrict__ verts,
                int* __restrict__ knnIdx, float* __restrict__ radii)
{
    __shared__ float d2chunk[16 * 256];   // 16 rows x 256 cols
    __shared__ float lvals[16 * 33];      // running 33 smallest d2 per row
    __shared__ int   lidxs[16 * 33];

    const int bx      = blockIdx.x;
    const int s       = bx / (NVERT / 16);
    const int rowTile = bx % (NVERT / 16);
    const int rowBase = rowTile * 16;
    const int tid     = threadIdx.x;
    const int lane    = tid & 31;
    const int w       = tid >> 5;
    const float* V    = verts + (size_t)s * NVERT * 3;

    for (int i = tid; i < 16 * 33; i += 256) { lvals[i] = BIGF; lidxs[i] = 0; }

    // A fragment (16x4 f32): lanes 0-15 hold K=0,1 ; lanes 16-31 hold K=2,3
    const int m  = rowBase + (lane & 15);
    const float ax = V[m * 3 + 0], ay = V[m * 3 + 1], az = V[m * 3 + 2];
    const float asq = ax * ax + ay * ay + az * az;
    v2f afrag;
    afrag.x = (lane < 16) ? ax : az;
    afrag.y = (lane < 16) ? ay : asq;

    float curmax0 = BIGF, curmax1 = BIGF;
    __syncthreads();

    const int NCHUNK = NVERT / 256;   // 24
    for (int chunk = 0; chunk < NCHUNK; ++chunk) {
        const int colChunkBase = chunk * 256;

        // ---- WMMA phase: 8 waves x 2 tiles = 16 tiles of 16 cols ----
#pragma unroll
        for (int tt = 0; tt < 2; ++tt) {
            const int tile = w * 2 + tt;
            const int n    = colChunkBase + tile * 16 + (lane & 15);
            const float bxc = V[n * 3 + 0], byc = V[n * 3 + 1], bzc = V[n * 3 + 2];
            const float bsq = bxc * bxc + byc * byc + bzc * bzc;
            v2f bfrag;
            bfrag.x = (lane < 16) ? (-2.0f * bxc) : (-2.0f * bzc);
            bfrag.y = (lane < 16) ? (-2.0f * byc) : 1.0f;
            v8f cfrag;
#pragma unroll
            for (int r = 0; r < 8; ++r) cfrag[r] = bsq;
            v8f dfrag = __builtin_amdgcn_wmma_f32_16x16x4_f32(
                false, afrag, false, bfrag, (short)0, cfrag, false, false);
            const int rowLo = (lane < 16) ? 0 : 8;
            const int col   = tile * 16 + (lane & 15);
#pragma unroll
            for (int r = 0; r < 8; ++r)
                d2chunk[(rowLo + r) * 256 + col] = fabsf(dfrag[r]);
        }
        // prefetch next chunk's column vertices (global_prefetch_b8)
        if (chunk + 1 < NCHUNK) {
            const float* pf = V + (size_t)(colChunkBase + 256) * 3 + (size_t)tid * 3;
            __builtin_prefetch(pf, 0, 1);
        }
        __syncthreads();

        // ---- selection phase: wave w owns rows 2w, 2w+1 ----
#pragma unroll
        for (int rr = 0; rr < 2; ++rr) {
            const int row = w * 2 + rr;
            float curmax  = rr ? curmax1 : curmax0;
            float* vals = &lvals[row * 33];
            int*   idxs = &lidxs[row * 33];
            for (int g = 0; g < 8; ++g) {
                const float v = d2chunk[row * 256 + g * 32 + lane];
                unsigned long long mask = __ballot(v < curmax);
                while (mask) {
                    const int src = __ffsll((unsigned long long)mask) - 1;
                    mask &= mask - 1ull;
                    const float cv = __shfl(v, src, 32);
                    const int   cj = colChunkBase + g * 32 + src;
                    // wave-wide argmax over 33 slots
                    float mv = vals[lane];
                    int   ms = lane;
#pragma unroll
                    for (int o = 16; o >= 1; o >>= 1) {
                        const float ov = __shfl_xor(mv, o, 32);
                        const int   os = __shfl_xor(ms, o, 32);
                        if (ov > mv || (ov == mv && os < ms)) { mv = ov; ms = os; }
                    }
                    const float v32 = vals[32];
                    if (v32 > mv) { mv = v32; ms = 32; }
                    if (cv < mv && lane == 0) { vals[ms] = cv; idxs[ms] = cj; }
                    curmax = mv;   // upper bound on post-insert 33rd smallest
                }
            }
            if (rr) curmax1 = curmax; else curmax0 = curmax;
        }
        __syncthreads();
    }

    // ---- finalize: radius = sqrt(33rd smallest); emit other 32 indices ----
    for (int rr = 0; rr < 2; ++rr) {
        const int row = w * 2 + rr;
        float* vals = &lvals[row * 33];
        int*   idxs = &lidxs[row * 33];
        float mv = vals[lane];
        int   ms = lane;
#pragma unroll
        for (int o = 16; o >= 1; o >>= 1) {
            const float ov = __shfl_xor(mv, o, 32);
            const int   os = __shfl_xor(ms, o, 32);
            if (ov > mv || (ov == mv && os < ms)) { mv = ov; ms = os; }
        }
        const float v32 = vals[32];
        if (v32 > mv) { mv = v32; ms = 32; }
        const int gr = s * NVERT + rowBase + row;
        if (lane == 0) radii[gr] = sqrtf(fmaxf(mv, EPSF));
        if (lane != ms) {
            const int pos = lane - (lane > ms ? 1 : 0);
            knnIdx[(size_t)gr * KNN + pos] = idxs[lane];
        }
        if (lane == 0 && ms != 32)
            knnIdx[(size_t)gr * KNN + 31] = idxs[32];
    }
}

// ---------------- 3x3 symmetric eigensolver helpers ----------------
__device__ __forceinline__ void evec_for(float a00, float a01, float a02,
                                         float a11, float a12, float a22,
                                         float lam, float out[3])
{
    const float r0x = a00 - lam, r0y = a01, r0z = a02;
    const float r1x = a01, r1y = a11 - lam, r1z = a12;
    const float r2x = a02, r2y = a12, r2z = a22 - lam;
    float c0x = r0y * r1z - r0z * r1y, c0y = r0z * r1x - r0x * r1z, c0z = r0x * r1y - r0y * r1x;
    float c1x = r0y * r2z - r0z * r2y, c1y = r0z * r2x - r0x * r2z, c1z = r0x * r2y - r0y * r2x;
    float c2x = r1y * r2z - r1z * r2y, c2y = r1z * r2x - r1x * r2z, c2z = r1x * r2y - r1y * r2x;
    const float n0 = c0x * c0x + c0y * c0y + c0z * c0z;
    const float n1 = c1x * c1x + c1y * c1y + c1z * c1z;
    const float n2 = c2x * c2x + c2y * c2y + c2z * c2z;
    float bx = c0x, by = c0y, bz = c0z, bn = n0;
    if (n1 > bn) { bx = c1x; by = c1y; bz = c1z; bn = n1; }
    if (n2 > bn) { bx = c2x; by = c2y; bz = c2z; bn = n2; }
    const float inv = rsqrtf(fmaxf(bn, 1e-30f));
    out[0] = bx * inv; out[1] = by * inv; out[2] = bz * inv;
}

// ---------------- Kernel 2: LRF + log-map + template barycentric ----------
// One wave per vertex; lane == neighbor slot (KNN == wave32 width).
__global__ __launch_bounds__(256)
void bary_kernel(const float* __restrict__ verts, const float* __restrict__ tmpl,
                 const int* __restrict__ knnIdx, const float* __restrict__ radii,
                 float* __restrict__ out)
{
    __shared__ float sproj[8][32][2];
    __shared__ float spidx[8][32];

    const int lane = threadIdx.x & 31;
    const int w    = threadIdx.x >> 5;
    const int vid  = blockIdx.x * 8 + w;          // 0 .. 2*NVERT-1
    const int s    = vid / NVERT;
    const int i    = vid - s * NVERT;
    const float* V = verts + (size_t)s * NVERT * 3;

    const float cx = V[i * 3 + 0], cy = V[i * 3 + 1], cz = V[i * 3 + 2];
    const int nb = knnIdx[(size_t)vid * KNN + lane];
    const float nx = V[nb * 3 + 0] - cx;
    const float ny = V[nb * 3 + 1] - cy;
    const float nz = V[nb * 3 + 2] - cz;
    const float radius = radii[vid];

    const float d   = sqrtf(fmaxf(nx * nx + ny * ny + nz * nz, EPSF));
    const float wgt = fmaxf(radius - d, 0.0f);

    // weighted covariance (wave reductions -> uniform values)
    const float wsum = wave_sum32(wgt);
    float c00 = wave_sum32(wgt * nx * nx);
    float c01 = wave_sum32(wgt * nx * ny);
    float c02 = wave_sum32(wgt * nx * nz);
    float c11 = wave_sum32(wgt * ny * ny);
    float c12 = wave_sum32(wgt * ny * nz);
    float c22 = wave_sum32(wgt * nz * nz);
    const float inv = 1.0f / fmaxf(wsum, 1e-30f);
    c00 *= inv; c01 *= inv; c02 *= inv; c11 *= inv; c12 *= inv; c22 *= inv;

    // eigenvalues: trigonometric closed form (largest & smallest)
    const float q  = (c00 + c11 + c22) * (1.0f / 3.0f);
    const float p1 = c01 * c01 + c02 * c02 + c12 * c12;
    const float d0 = c00 - q, d1 = c11 - q, d2c = c22 - q;
    const float p2 = d0 * d0 + d1 * d1 + d2c * d2c + 2.0f * p1;
    const float p  = sqrtf(fmaxf(p2, 1e-30f) * (1.0f / 6.0f));
    const float ip = 1.0f / fmaxf(p, 1e-30f);
    const float b00 = d0 * ip, b01 = c01 * ip, b02 = c02 * ip;
    const float b11 = d1 * ip, b12 = c12 * ip, b22 = d2c * ip;
    float detB = b00 * (b11 * b22 - b12 * b12)
               - b01 * (b01 * b22 - b12 * b02)
               + b02 * (b01 * b12 - b11 * b02);
    float r = 0.5f * detB;
    r = fminf(1.0f, fmaxf(-1.0f, r));
    const float phi = acosf(r) * (1.0f / 3.0f);
    const float lamBig   = q + 2.0f * p * cosf(phi);
    const float lamSmall = q + 2.0f * p * cosf(phi + 2.0943951023931953f);

    float xv[3], zv[3];
    evec_for(c00, c01, c02, c11, c12, c22, lamBig, xv);
    evec_for(c00, c01, c02, c11, c12, c22, lamSmall, zv);

    // sign disambiguation via ballot/popcount over the 32 neighbors
    {
        const float dotx = nx * xv[0] + ny * xv[1] + nz * xv[2];
        const int pos = __popcll(__ballot(dotx >= 0.0f));
        const float sg = (2 * pos >= KNN) ? 1.0f : -1.0f;
        xv[0] *= sg; xv[1] *= sg; xv[2] *= sg;
    }
    {
        const float dotz = nx * zv[0] + ny * zv[1] + nz * zv[2];
        const int pos = __popcll(__ballot(dotz >= 0.0f));
        const float sg = (2 * pos >= KNN) ? 1.0f : -1.0f;
        zv[0] *= sg; zv[1] *= sg; zv[2] *= sg;
    }
    const float yv0 = zv[1] * xv[2] - zv[2] * xv[1];
    const float yv1 = zv[2] * xv[0] - zv[0] * xv[2];
    const float yv2 = zv[0] * xv[1] - zv[1] * xv[0];

    // log-map projection
    const float u  = nx * xv[0] + ny * xv[1] + nz * xv[2];
    const float vq = nx * yv0 + ny * yv1 + nz * yv2;
    const float n2 = sqrtf(fmaxf(u * u + vq * vq, EPSF));
    const float scale = (n2 > 1e-6f) ? (d / n2) : 0.0f;
    sproj[w][lane][0] = u * scale;
    sproj[w][lane][1] = vq * scale;
    spidx[w][lane]    = (float)nb;
    __syncthreads();

    // lanes switch role: template points (40 = 32 + 8)
#pragma unroll
    for (int it = 0; it < 2; ++it) {
        const int t = it * 32 + lane;
        if (t < NTPL) {
            const float tx = tmpl[2 * t + 0], ty = tmpl[2 * t + 1];
            float b0 = BIGF, b1 = BIGF, b2 = BIGF;
            int i0 = 0, i1 = 0, i2 = 0;
            for (int k = 0; k < KNN; ++k) {
                const float dx = tx - sproj[w][k][0];
                const float dy = ty - sproj[w][k][1];
                const float dd = dx * dx + dy * dy;
                if (dd < b0)      { b2 = b1; i2 = i1; b1 = b0; i1 = i0; b0 = dd; i0 = k; }
                else if (dd < b1) { b2 = b1; i2 = i1; b1 = dd; i1 = k; }
                else if (dd < b2) { b2 = dd; i2 = k; }
            }
            const float p0x = sproj[w][i0][0], p0y = sproj[w][i0][1];
            const float p1x = sproj[w][i1][0], p1y = sproj[w][i1][1];
            const float p2x = sproj[w][i2][0], p2y = sproj[w][i2][1];
            const float v0x = p2x - p0x, v0y = p2y - p0y;
            const float v1x = p1x - p0x, v1y = p1y - p0y;
            const float v2x = tx - p0x,  v2y = ty - p0y;
            const float dot00 = v0x * v0x + v0y * v0y;
            const float dot01 = v0x * v1x + v0y * v1y;
            const float dot02 = v0x * v2x + v0y * v2y;
            const float dot11 = v1x * v1x + v1y * v1y;
            const float dot12 = v1x * v2x + v1y * v2y;
            const float denom = dot00 * dot11 - dot01 * dot01;
            const float w2 = (dot11 * dot02 - dot01 * dot12) / denom;
            const float w1 = (dot00 * dot12 - dot01 * dot02) / denom;
            const float w0 = 1.0f - w2 - w1;
            const size_t base = ((size_t)vid * NTPL + t) * 6;
            out[base + 0] = spidx[w][i0]; out[base + 1] = w2;
            out[base + 2] = spidx[w][i1]; out[base + 3] = w1;
            out[base + 4] = spidx[w][i2]; out[base + 5] = w0;
        }
    }
}

extern "C" void kernel_launch(void* const* d_in, const int* in_sizes, int n_in,
                              void* d_out, int out_size, void* d_ws, size_t ws_size,
                              hipStream_t stream) {
    const float* verts = (const float*)d_in[0];   // (2, 6144, 3) f32
    const float* tmpl  = (const float*)d_in[1];   // (5, 8, 2) f32
    // d_in[2] = n_neighbors == 32 (hardwired: lane == neighbor under wave32)

    int*   knnIdx = (int*)d_ws;                                     // 2*6144*32 ints
    float* radii  = (float*)((char*)d_ws + (size_t)2 * NVERT * KNN * sizeof(int));
    float* out    = (float*)d_out;

    knn_kernel<<<2 * (NVERT / 16), 256, 0, stream>>>(verts, knnIdx, radii);
    bary_kernel<<<2 * NVERT / 8, 256, 0, stream>>>(verts, tmpl, knnIdx, radii, out);
}